// Net_46273977647788
// MI455X (gfx1250) — compile-verified
//
#include <hip/hip_runtime.h>

typedef float v2f __attribute__((ext_vector_type(2)));
typedef float v8f __attribute__((ext_vector_type(8)));

#define DH 128   // hidden dim
#define DO 16    // output dim

// ---------------- degree / norm ----------------
__global__ void k_deg_init(float* __restrict__ deg, int N) {
  int i = blockIdx.x * blockDim.x + threadIdx.x;
  if (i < N) deg[i] = 1.0f;  // self-loop contributes 1
}
__global__ void k_deg_accum(const int* __restrict__ dst, float* __restrict__ deg, int E) {
  int e = blockIdx.x * blockDim.x + threadIdx.x;
  if (e < E) atomicAdd(&deg[dst[e]], 1.0f);
}
__global__ void k_dinv(float* __restrict__ deg, int N) {
  int i = blockIdx.x * blockDim.x + threadIdx.x;
  if (i < N) { float d = deg[i]; deg[i] = d > 0.0f ? rsqrtf(d) : 0.0f; }
}

// ---------------- GEMM1: G = (emb[x] @ W1) * dinv[row], ACC = G ----------------
// One wave per 16x16 output tile, f32 WMMA 16x16x4, K=128 in 32 steps.
__global__ void k_gemm1(const int* __restrict__ x, const float* __restrict__ emb,
                        const float* __restrict__ W1, const float* __restrict__ dinv,
                        float* __restrict__ G, float* __restrict__ ACC, int N) {
  const int lane = threadIdx.x;                 // 0..31 (wave32)
  const int tilesN = DH / 16;                   // 8
  int tile = blockIdx.x * blockDim.y + threadIdx.y;
  int mt = (N + 15) >> 4;
  if (tile >= mt * tilesN) return;
  int tm = tile / tilesN;
  int tn = tile - tm * tilesN;

  int mrow = tm * 16 + (lane & 15);             // A row for this lane
  int n    = tn * 16 + (lane & 15);             // B/C column for this lane
  int koff = (lane >> 4) << 1;                  // lanes 16-31 hold K+2,K+3

  int node = x[mrow < N ? mrow : N - 1];
  const float* arow = emb + (size_t)node * DH;
  const float* bcol = W1 + n;

  v8f c = {0.f, 0.f, 0.f, 0.f, 0.f, 0.f, 0.f, 0.f};
#pragma unroll 8
  for (int k0 = 0; k0 < DH; k0 += 4) {
    v2f a = *(const v2f*)(arow + k0 + koff);    // contiguous -> b64 load
    v2f b;
    b.x = bcol[(size_t)(k0 + koff) * DH];
    b.y = bcol[(size_t)(k0 + koff + 1) * DH];
    c = __builtin_amdgcn_wmma_f32_16x16x4_f32(false, a, false, b, (short)0, c, false, false);
  }

  int rbase = tm * 16 + ((lane >> 4) << 3);     // VGPR r: M = r (+8 for upper lanes)
  if (tm * 16 + 16 <= N) {                      // wave-uniform: full tile, no guards
    float4 d0 = *(const float4*)(dinv + rbase);
    float4 d1 = *(const float4*)(dinv + rbase + 4);
    float sc[8] = {d0.x, d0.y, d0.z, d0.w, d1.x, d1.y, d1.z, d1.w};
    size_t idx = (size_t)rbase * DH + n;
#pragma unroll
    for (int r = 0; r < 8; ++r, idx += DH) {
      float v = c[r] * sc[r];
      G[idx] = v;
      ACC[idx] = v;                              // self-loop init
    }
  } else {                                       // remainder tile (at most one)
#pragma unroll
    for (int r = 0; r < 8; ++r) {
      int row = rbase + r;
      if (row < N) {
        float v = c[r] * dinv[row];
        size_t idx = (size_t)row * DH + n;
        G[idx] = v;
        ACC[idx] = v;
      }
    }
  }
}

// ---------------- GEMM2: G2 = (H @ W2) * dinv[row], ACC2 = G2 ----------------
__global__ void k_gemm2(const float* __restrict__ H, const float* __restrict__ W2,
                        const float* __restrict__ dinv,
                        float* __restrict__ G2, float* __restrict__ ACC2, int N) {
  const int lane = threadIdx.x;
  int tm = blockIdx.x * blockDim.y + threadIdx.y;
  if (tm >= ((N + 15) >> 4)) return;

  int mrow = tm * 16 + (lane & 15);
  int n    = lane & 15;
  int koff = (lane >> 4) << 1;

  const float* arow = H + (size_t)(mrow < N ? mrow : N - 1) * DH;
  const float* bcol = W2 + n;

  v8f c = {0.f, 0.f, 0.f, 0.f, 0.f, 0.f, 0.f, 0.f};
#pragma unroll 8
  for (int k0 = 0; k0 < DH; k0 += 4) {
    v2f a = *(const v2f*)(arow + k0 + koff);
    v2f b;
    b.x = bcol[(size_t)(k0 + koff) * DO];
    b.y = bcol[(size_t)(k0 + koff + 1) * DO];
    c = __builtin_amdgcn_wmma_f32_16x16x4_f32(false, a, false, b, (short)0, c, false, false);
  }

  int rbase = tm * 16 + ((lane >> 4) << 3);
  if (tm * 16 + 16 <= N) {                      // wave-uniform fast path
    float4 d0 = *(const float4*)(dinv + rbase);
    float4 d1 = *(const float4*)(dinv + rbase + 4);
    float sc[8] = {d0.x, d0.y, d0.z, d0.w, d1.x, d1.y, d1.z, d1.w};
    size_t idx = (size_t)rbase * DO + n;
#pragma unroll
    for (int r = 0; r < 8; ++r, idx += DO) {
      float v = c[r] * sc[r];
      G2[idx] = v;
      ACC2[idx] = v;
    }
  } else {
#pragma unroll
    for (int r = 0; r < 8; ++r) {
      int row = rbase + r;
      if (row < N) {
        float v = c[r] * dinv[row];
        size_t idx = (size_t)row * DO + n;
        G2[idx] = v;
        ACC2[idx] = v;
      }
    }
  }
}

// ---------------- edge aggregation: ACC[dst] += G[src] ----------------
// tpe = 2^shift threads per edge, each handling 4 consecutive floats.
__global__ void k_agg(const float* __restrict__ G, const int* __restrict__ src,
                      const int* __restrict__ dst, float* __restrict__ ACC,
                      int E, int rowF, int shift) {
  long long gid = (long long)blockIdx.x * blockDim.x + threadIdx.x;
  long long e = gid >> shift;
  if (e >= E) return;
  int q = ((int)(gid & ((1 << shift) - 1))) << 2;
  int s = src[e], d = dst[e];
  float4 v = *(const float4*)(G + (size_t)s * rowF + q);
  float* p = ACC + (size_t)d * rowF + q;
  atomicAdd(p + 0, v.x);
  atomicAdd(p + 1, v.y);
  atomicAdd(p + 2, v.z);
  atomicAdd(p + 3, v.w);
}

// ---------------- finalize conv1 + ReLU + LayerNorm (wave per row) ----------------
__global__ void k_relu_ln(const float* __restrict__ ACC, const float* __restrict__ dinv,
                          const float* __restrict__ b1, const float* __restrict__ lng,
                          const float* __restrict__ lnb, float* __restrict__ H, int N) {
  int lane = threadIdx.x;                        // 0..31, 4 floats each
  int row = blockIdx.x * blockDim.y + threadIdx.y;
  if (row >= N) return;
  float s = dinv[row];
  float4 v  = ((const float4*)(ACC + (size_t)row * DH))[lane];
  float4 bb = ((const float4*)b1)[lane];
  v.x = fmaxf(v.x * s + bb.x, 0.0f);
  v.y = fmaxf(v.y * s + bb.y, 0.0f);
  v.z = fmaxf(v.z * s + bb.z, 0.0f);
  v.w = fmaxf(v.w * s + bb.w, 0.0f);
  float sum = v.x + v.y + v.z + v.w;
  float sq  = v.x * v.x + v.y * v.y + v.z * v.z + v.w * v.w;
#pragma unroll
  for (int off = 16; off >= 1; off >>= 1) {
    sum += __shfl_xor(sum, off, 32);
    sq  += __shfl_xor(sq,  off, 32);
  }
  float mu   = sum * (1.0f / DH);
  float var  = sq * (1.0f / DH) - mu * mu;
  float rstd = rsqrtf(var + 1e-5f);
  float4 gg = ((const float4*)lng)[lane];
  float4 be = ((const float4*)lnb)[lane];
  float4 o;
  o.x = (v.x - mu) * rstd * gg.x + be.x;
  o.y = (v.y - mu) * rstd * gg.y + be.y;
  o.z = (v.z - mu) * rstd * gg.z + be.z;
  o.w = (v.w - mu) * rstd * gg.w + be.w;
  ((float4*)(H + (size_t)row * DH))[lane] = o;
}

// ---------------- z = ACC2 * dinv[row] + b2 ----------------
__global__ void k_z(const float* __restrict__ ACC2, const float* __restrict__ dinv,
                    const float* __restrict__ b2, float* __restrict__ z, int N) {
  int gid = blockIdx.x * blockDim.x + threadIdx.x;
  if (gid < N * DO) {
    int row = gid >> 4;
    int j = gid & 15;
    z[gid] = ACC2[gid] * dinv[row] + b2[j];
  }
}

// ---------------- decode: out = ((concat(z[a],z[b]) @ L1) + bl1) @ L2 + bl2 ----------------
__global__ void k_decode(const float* __restrict__ z, const int* __restrict__ eli,
                         int EL, const float* __restrict__ L1, const float* __restrict__ bl1,
                         const float* __restrict__ L2, const float* __restrict__ bl2,
                         float* __restrict__ out) {
  __shared__ float sL1[32 * 16];
  __shared__ float sb1[16];
  __shared__ float sL2[16];
  __shared__ float sb2;
  for (int i = threadIdx.x; i < 512; i += blockDim.x) sL1[i] = L1[i];
  if (threadIdx.x < 16) { sb1[threadIdx.x] = bl1[threadIdx.x]; sL2[threadIdx.x] = L2[threadIdx.x]; }
  if (threadIdx.x == 0) sb2 = bl2[0];
  __syncthreads();

  int e = blockIdx.x * blockDim.x + threadIdx.x;
  if (e >= EL) return;
  int a = eli[e];
  int b = eli[EL + e];
  float feat[32];
  const float4* za = (const float4*)(z + (size_t)a * DO);
  const float4* zb = (const float4*)(z + (size_t)b * DO);
#pragma unroll
  for (int q = 0; q < 4; ++q) {
    float4 t = za[q];
    feat[q * 4 + 0] = t.x; feat[q * 4 + 1] = t.y; feat[q * 4 + 2] = t.z; feat[q * 4 + 3] = t.w;
  }
#pragma unroll
  for (int q = 0; q < 4; ++q) {
    float4 t = zb[q];
    feat[16 + q * 4 + 0] = t.x; feat[16 + q * 4 + 1] = t.y; feat[16 + q * 4 + 2] = t.z; feat[16 + q * 4 + 3] = t.w;
  }
  float o = sb2;
#pragma unroll 4
  for (int j = 0; j < 16; ++j) {
    float h = sb1[j];
#pragma unroll
    for (int i = 0; i < 32; ++i) h += feat[i] * sL1[i * 16 + j];
    o += h * sL2[j];
  }
  out[e] = o;
}

extern "C" void kernel_launch(void* const* d_in, const int* in_sizes, int n_in,
                              void* d_out, int out_size, void* d_ws, size_t ws_size,
                              hipStream_t stream) {
  (void)n_in; (void)out_size; (void)ws_size;
  const int*   x    = (const int*)d_in[0];
  const int*   ei   = (const int*)d_in[1];
  const int*   eli  = (const int*)d_in[2];
  const float* emb  = (const float*)d_in[3];
  const float* W1   = (const float*)d_in[4];
  const float* b1   = (const float*)d_in[5];
  const float* lng  = (const float*)d_in[6];
  const float* lnb  = (const float*)d_in[7];
  const float* W2   = (const float*)d_in[8];
  const float* b2   = (const float*)d_in[9];
  const float* L1   = (const float*)d_in[10];
  const float* bl1  = (const float*)d_in[11];
  const float* L2   = (const float*)d_in[12];
  const float* bl2  = (const float*)d_in[13];

  const int N  = in_sizes[0];
  const int E  = in_sizes[1] / 2;
  const int EL = in_sizes[2] / 2;
  const int* src = ei;
  const int* dst = ei + E;

  // workspace layout (every byte written before read, each call)
  char* w = (char*)d_ws;
  size_t off = 0;
  auto take = [&](size_t bytes) -> void* {
    void* p = w + off;
    off += (bytes + 255) & ~(size_t)255;
    return p;
  };
  float* deg = (float*)take((size_t)N * 4);        // deg -> dinv (in place)
  float* A   = (float*)take((size_t)N * DH * 4);   // G (msgs), later H (post-LN)
  float* B   = (float*)take((size_t)N * DH * 4);   // ACC (conv1 accumulator)
  float* C   = (float*)take((size_t)N * DO * 4);   // G2, later z
  float* D   = (float*)take((size_t)N * DO * 4);   // ACC2

  // 1) degrees + dinv
  k_deg_init<<<(N + 255) / 256, 256, 0, stream>>>(deg, N);
  k_deg_accum<<<(E + 255) / 256, 256, 0, stream>>>(dst, deg, E);
  k_dinv<<<(N + 255) / 256, 256, 0, stream>>>(deg, N);

  const int mt = (N + 15) / 16;

  // 2) conv1 GEMM (WMMA f32): G = (emb[x] @ W1) * dinv, ACC = G
  {
    dim3 blk(32, 8);
    int tiles = mt * (DH / 16);
    k_gemm1<<<(tiles + 7) / 8, blk, 0, stream>>>(x, emb, W1, deg, A, B, N);
  }
  // 3) edge scatter-add, 128 floats/row, 32 threads/edge
  {
    long long tot = (long long)E * 32;
    k_agg<<<(unsigned)((tot + 255) / 256), 256, 0, stream>>>(A, src, dst, B, E, DH, 5);
  }
  // 4) finalize + ReLU + LayerNorm -> H (reuses A)
  {
    dim3 blk(32, 8);
    k_relu_ln<<<(N + 7) / 8, blk, 0, stream>>>(B, deg, b1, lng, lnb, A, N);
  }
  // 5) conv2 GEMM (WMMA f32): G2 = (H @ W2) * dinv, ACC2 = G2
  {
    dim3 blk(32, 8);
    k_gemm2<<<(mt + 7) / 8, blk, 0, stream>>>(A, W2, deg, C, D, N);
  }
  // 6) edge scatter-add, 16 floats/row, 4 threads/edge
  {
    long long tot = (long long)E * 4;
    k_agg<<<(unsigned)((tot + 255) / 256), 256, 0, stream>>>(C, src, dst, D, E, DO, 2);
  }
  // 7) z = ACC2 * dinv + b2 (reuses C)
  k_z<<<(N * DO + 255) / 256, 256, 0, stream>>>(D, deg, b2, C, N);
  // 8) decode MLP
  k_decode<<<(EL + 255) / 256, 256, 0, stream>>>(C, eli, EL, L1, bl1, L2, bl2, (float*)d_out);
}